// GNNModel_2680059593194
// MI455X (gfx1250) — compile-verified
//
#include <hip/hip_runtime.h>

#define H 64   // hidden dim (fixed by reference)

typedef __attribute__((ext_vector_type(2))) float v2f;
typedef __attribute__((ext_vector_type(8))) float v8f;

// ---------- degree / norm ----------
__global__ void init_deg_kernel(float* deg, int n) {
    int i = blockIdx.x * blockDim.x + threadIdx.x;
    if (i < n) deg[i] = 1.0f;  // self-loop
}

__global__ void count_deg_kernel(const int* __restrict__ dst, float* deg, int e) {
    int i = blockIdx.x * blockDim.x + threadIdx.x;
    if (i < e) atomicAdd(&deg[dst[i]], 1.0f);
}

__global__ void dinv_zero_t_kernel(const float* __restrict__ deg,
                                   float* dinv, float* t, int n) {
    int i = blockIdx.x * blockDim.x + threadIdx.x;
    if (i < n) { dinv[i] = rsqrtf(deg[i]); t[i] = 0.0f; }
}

__global__ void zero_acc_kernel(float4* acc, int n4) {
    int i = blockIdx.x * blockDim.x + threadIdx.x;
    if (i < n4) acc[i] = make_float4(0.f, 0.f, 0.f, 0.f);
}

// ---------- layer 1 (feature dim 1 -> scalar aggregation) ----------
__global__ void l1_edge_kernel(const int* __restrict__ src, const int* __restrict__ dst,
                               const float* __restrict__ x, const float* __restrict__ dinv,
                               float* t, int e) {
    int i = blockIdx.x * blockDim.x + threadIdx.x;
    if (i < e) {
        int s = src[i];
        atomicAdd(&t[dst[i]], dinv[s] * x[s]);
    }
}

__global__ void l1_node_kernel(const float* __restrict__ x, const float* __restrict__ dinv,
                               const float* __restrict__ t, const float* __restrict__ W1,
                               const float* __restrict__ b1, float* h1, int n) {
    int tid = blockIdx.x * blockDim.x + threadIdx.x;
    int node = tid >> 4, q = tid & 15;     // 16 threads/node, float4 each
    if (node >= n) return;
    float di = dinv[node];
    float c = di * (t[node] + di * x[node]);
    float4 w = ((const float4*)W1)[q];
    float4 bb = ((const float4*)b1)[q];
    float4 r;
    r.x = fmaxf(c * w.x + bb.x, 0.f);
    r.y = fmaxf(c * w.y + bb.y, 0.f);
    r.z = fmaxf(c * w.z + bb.z, 0.f);
    r.w = fmaxf(c * w.w + bb.w, 0.f);
    ((float4*)h1)[node * 16 + q] = r;
}

// ---------- layer 2 GEMM: g = h1 @ W2  via V_WMMA_F32_16X16X4_F32 ----------
// Block = 256 threads = 8 waves; each wave owns one 16x16 output tile.
// Block covers 32 rows x 64 cols (2 row-tiles x 4 col-tiles).
__global__ void __launch_bounds__(256)
gemm_wmma_kernel(const float* __restrict__ A,   // h1 [n,64]
                 const float* __restrict__ B,   // W2 [64,64]
                 float* __restrict__ Cm,        // g  [n,64]
                 int n) {
    int wid  = threadIdx.x >> 5;
    int lane = threadIdx.x & 31;
    int m0 = blockIdx.x * 32 + (wid >> 2) * 16;
    int n0 = (wid & 3) * 16;
    if (m0 >= n) return;                    // whole-wave skip keeps EXEC uniform

    int half = lane >> 4;                   // 0: lanes 0-15, 1: lanes 16-31
    int l15  = lane & 15;
    int kb   = half * 2;                    // K sub-offset held by this half-wave

    int arow = m0 + l15;                    // A-matrix row for this lane
    if (arow >= n - 1) arow = n - 1;        // clamp (dup last row; stores predicated)
    const float* Ap = A + (size_t)arow * H;

    v8f c = {0.f, 0.f, 0.f, 0.f, 0.f, 0.f, 0.f, 0.f};
#pragma unroll
    for (int k = 0; k < H; k += 4) {
        v2f a, b;
        a.x = Ap[k + kb];
        a.y = Ap[k + kb + 1];
        b.x = B[(k + kb) * H + n0 + l15];
        b.y = B[(k + kb + 1) * H + n0 + l15];
        // D = A(16x4) x B(4x16) + C, pure fp32
        c = __builtin_amdgcn_wmma_f32_16x16x4_f32(
                /*neg_a=*/false, a, /*neg_b=*/false, b,
                /*c_mod=*/(short)0, c, /*reuse_a=*/false, /*reuse_b=*/false);
    }

#pragma unroll
    for (int r = 0; r < 8; ++r) {
        int row = m0 + r + half * 8;        // C/D layout: VGPR r -> M = r + 8*half
        if (row < n) Cm[(size_t)row * H + n0 + l15] = c[r];
    }
}

// ---------- layer 2 propagation ----------
__global__ void l2_edge_kernel(const int* __restrict__ src, const int* __restrict__ dst,
                               const float* __restrict__ dinv, const float* __restrict__ g,
                               float* acc, int e) {
    long long tid = (long long)blockIdx.x * blockDim.x + threadIdx.x;
    int eidx = (int)(tid >> 4);
    if (eidx >= e) return;
    int q = (int)(tid & 15);
    int s = src[eidx], d = dst[eidx];
    float w = dinv[s];
    float4 gv = ((const float4*)g)[s * 16 + q];
    float* out = acc + (size_t)d * H + q * 4;
    atomicAdd(out + 0, w * gv.x);
    atomicAdd(out + 1, w * gv.y);
    atomicAdd(out + 2, w * gv.z);
    atomicAdd(out + 3, w * gv.w);
}

__global__ void l2_node_kernel(const float* __restrict__ dinv, const float* __restrict__ acc,
                               const float* __restrict__ g, const float* __restrict__ b2,
                               float* h2, int n) {
    int tid = blockIdx.x * blockDim.x + threadIdx.x;
    int node = tid >> 4, q = tid & 15;
    if (node >= n) return;
    float di = dinv[node];
    float di2 = di * di;
    float4 av = ((const float4*)acc)[node * 16 + q];
    float4 gv = ((const float4*)g)[node * 16 + q];
    float4 bb = ((const float4*)b2)[q];
    float4 r;
    r.x = fmaxf(di * av.x + di2 * gv.x + bb.x, 0.f);
    r.y = fmaxf(di * av.y + di2 * gv.y + bb.y, 0.f);
    r.z = fmaxf(di * av.z + di2 * gv.z + bb.z, 0.f);
    r.w = fmaxf(di * av.w + di2 * gv.w + bb.w, 0.f);
    ((float4*)h2)[node * 16 + q] = r;
}

// ---------- head: out = h2 @ Wh + bh ----------
__global__ void head_kernel(const float* __restrict__ h2, const float* __restrict__ Wh,
                            const float* __restrict__ bh, float* out, int n) {
    int i = blockIdx.x * blockDim.x + threadIdx.x;
    if (i >= n) return;
    const float4* hp = (const float4*)(h2 + (size_t)i * H);
    const float4* wp = (const float4*)Wh;
    float s = 0.f;
#pragma unroll
    for (int q = 0; q < 16; ++q) {
        float4 h = hp[q], w = wp[q];
        s += h.x * w.x + h.y * w.y + h.z * w.z + h.w * w.w;
    }
    out[i] = s + bh[0];
}

extern "C" void kernel_launch(void* const* d_in, const int* in_sizes, int n_in,
                              void* d_out, int out_size, void* d_ws, size_t ws_size,
                              hipStream_t stream) {
    (void)n_in; (void)out_size; (void)ws_size;
    const float* x  = (const float*)d_in[0];
    const int*   ei = (const int*)d_in[1];
    const float* W1 = (const float*)d_in[2];
    const float* b1 = (const float*)d_in[3];
    const float* W2 = (const float*)d_in[4];
    const float* b2 = (const float*)d_in[5];
    const float* Wh = (const float*)d_in[6];
    const float* bh = (const float*)d_in[7];
    float* out = (float*)d_out;

    const int n = in_sizes[0];          // x is [N,1]
    const int e = in_sizes[1] / 2;      // edge_index is [2,E]
    const int* src = ei;
    const int* dst = ei + e;

    float* ws   = (float*)d_ws;
    float* dinv = ws;                            // n
    float* t    = ws + (size_t)n;                // n
    float* h1   = ws + 2 * (size_t)n;            // 64n (reused as h2 later)
    float* g    = h1 + (size_t)H * n;            // 64n
    float* acc  = g  + (size_t)H * n;            // 64n
    float* deg  = g;                             // alias: dead before GEMM writes g
    float* h2   = h1;                            // alias: h1 dead after GEMM

    const int B = 256;
    int nb_n   = (n + B - 1) / B;
    int nb_e   = (e + B - 1) / B;
    int nb_n16 = (int)(((size_t)n * 16 + B - 1) / B);
    int nb_e16 = (int)(((size_t)e * 16 + B - 1) / B);
    int nb_acc = (int)(((size_t)n * 16 + B - 1) / B);   // 64n floats = 16n float4
    int nb_mm  = (n + 31) / 32;                          // 32 rows per block

    init_deg_kernel<<<nb_n, B, 0, stream>>>(deg, n);
    count_deg_kernel<<<nb_e, B, 0, stream>>>(dst, deg, e);
    dinv_zero_t_kernel<<<nb_n, B, 0, stream>>>(deg, dinv, t, n);
    zero_acc_kernel<<<nb_acc, B, 0, stream>>>((float4*)acc, n * 16);

    // layer 1
    l1_edge_kernel<<<nb_e, B, 0, stream>>>(src, dst, x, dinv, t, e);
    l1_node_kernel<<<nb_n16, B, 0, stream>>>(x, dinv, t, W1, b1, h1, n);

    // layer 2: dense transform on tensor cores, then scatter-aggregate
    gemm_wmma_kernel<<<nb_mm, B, 0, stream>>>(h1, W2, g, n);
    l2_edge_kernel<<<nb_e16, B, 0, stream>>>(src, dst, dinv, g, acc, e);
    l2_node_kernel<<<nb_n16, B, 0, stream>>>(dinv, acc, g, b2, h2, n);

    // head
    head_kernel<<<nb_n, B, 0, stream>>>(h2, Wh, bh, out, n);
}